// DigitCaps_Layer_12678743458055
// MI455X (gfx1250) — compile-verified
//
#include <hip/hip_runtime.h>
#include <math.h>

typedef __attribute__((ext_vector_type(2))) float v2f;
typedef __attribute__((ext_vector_type(8))) float v8f;

#define IN_CAP_N 1152
#define IN_CAP_SZ 5
#define OUT_CAP_N 55
#define BATCH     8192
#define ROWS      (BATCH * IN_CAP_SZ)   // 40960 rows of length 1152
#define KSPLIT    4
#define KCHUNK    (IN_CAP_N / KSPLIT)   // 288 floats per wave (72 WMMA k-steps)

// ---------------------------------------------------------------------------
// Kernel 1: u_hat[row] = dot(u[row, 0:1152], W[0:1152])   (row = b*5 + i)
//
// WMMA f32 16x16x4 per wave32:
//   A[m,k] = W[k]  (replicated over all 16 rows m, staged in LDS)
//   B[k,n] = u[row_base + n, k]
//   => D[m,n] = dot(W_chunk, u_row_n), identical for all m.
// f32 operand layout (ISA 7.12.2): lane L supplies k = r + 2*(L>=16) for
// VGPR r in {0,1}; B/D column n = L&15 -> lane L's acc[0] is the dot of
// row_base + (L&15).
//
// Latency hiding: 4 waves split K (288 floats each) for one 16-row tile;
// a 256-thread block = 8 waves = 2 tiles. 1280 blocks -> 10240 waves
// (~10 waves/SIMD) so ~10 KB of loads in flight per SIMD. Partial dots are
// reduced with a 4-term LDS sum. u loads are non-temporal (189 MB single-use
// stream ~ L2 capacity).
// ---------------------------------------------------------------------------
__global__ __launch_bounds__(256) void uhat_wmma_kernel(
    const float* __restrict__ u,      // (ROWS, 1152) contiguous
    const float* __restrict__ W,      // (1152)
    float* __restrict__ u_hat)        // (ROWS)
{
    __shared__ float wlds[IN_CAP_N];
    __shared__ float part[2][KSPLIT][16];
    for (int t = threadIdx.x; t < IN_CAP_N; t += 256) wlds[t] = W[t];
    __syncthreads();

    const int lane = threadIdx.x & 31;
    const int wave = threadIdx.x >> 5;
    const int tl   = wave >> 2;                 // tile-in-block: 0..1
    const int kc   = wave & (KSPLIT - 1);       // K-chunk: 0..3
    const int row_base = (blockIdx.x * 2 + tl) * 16;
    const int n  = lane & 15;                   // B-matrix column -> u row
    const int hi = lane >> 4;                   // upper half-wave: k += 2

    const float* urow = u + (size_t)(row_base + n) * IN_CAP_N + kc * KCHUNK + 2 * hi;
    const float* wrow = &wlds[kc * KCHUNK + 2 * hi];

    v8f acc = {0.f, 0.f, 0.f, 0.f, 0.f, 0.f, 0.f, 0.f};

    #pragma unroll 8
    for (int k = 0; k < KCHUNK; k += 4) {
        v2f a = *(const v2f*)(wrow + k);                          // W from LDS
        v2f b = __builtin_nontemporal_load((const v2f*)(urow + k)); // u stream
        acc = __builtin_amdgcn_wmma_f32_16x16x4_f32(
            /*neg_a=*/false, a, /*neg_b=*/false, b,
            /*c_mod=*/(short)0, acc, /*reuse_a=*/false, /*reuse_b=*/false);
    }

    // lane L (<16) holds this K-chunk's partial dot for row_base + L
    if (lane < 16) part[tl][kc][lane] = acc[0];
    __syncthreads();

    if (threadIdx.x < 32) {
        const int t2 = threadIdx.x >> 4;        // tile-in-block
        const int r  = threadIdx.x & 15;        // row-in-tile
        float s = part[t2][0][r] + part[t2][1][r] + part[t2][2][r] + part[t2][3][r];
        u_hat[(blockIdx.x * 2 + t2) * 16 + r] = s;
    }
}

// ---------------------------------------------------------------------------
// Kernel 2: dynamic routing (2 iterations), one thread per batch element.
//
// Iter 1 collapses: all 5 logit columns equal b_param, so
//   s_j = softmax(b)_j * (sum_i u_hat_i),  v_j = squash(s_j).
// Iter 2: per column i, softmax_j(b[j] + v[j]*uh[i]); s_j += c_ji * uh[i].
// Output v = squash(s) of iteration 2.  squash(s) = |s|/(1+s^2) * s.
// ---------------------------------------------------------------------------
__global__ __launch_bounds__(64) void routing_kernel(
    const float* __restrict__ u_hat,   // (BATCH, 5)
    const float* __restrict__ bparam,  // (55)
    float* __restrict__ out)           // (BATCH, 55)
{
    __shared__ float bp[OUT_CAP_N];
    if (threadIdx.x < OUT_CAP_N) bp[threadIdx.x] = bparam[threadIdx.x];
    __syncthreads();

    const int b = blockIdx.x * 64 + threadIdx.x;

    float uh[IN_CAP_SZ];
    #pragma unroll
    for (int i = 0; i < IN_CAP_SZ; ++i) uh[i] = u_hat[b * IN_CAP_SZ + i];

    // ---- iteration 1 (all columns identical) ----
    float mx = -INFINITY;
    #pragma unroll
    for (int j = 0; j < OUT_CAP_N; ++j) mx = fmaxf(mx, bp[j]);
    float den = 0.f;
    #pragma unroll
    for (int j = 0; j < OUT_CAP_N; ++j) den += __expf(bp[j] - mx);
    const float inv = __frcp_rn(den);

    float S = 0.f;
    #pragma unroll
    for (int i = 0; i < IN_CAP_SZ; ++i) S += uh[i];

    float v[OUT_CAP_N];
    #pragma unroll
    for (int j = 0; j < OUT_CAP_N; ++j) {
        float s  = __expf(bp[j] - mx) * inv * S;
        float s2 = s * s;
        v[j] = fabsf(s) / (1.f + s2) * s;       // squash (scalar capsule)
    }

    // ---- iteration 2 ----
    float sacc[OUT_CAP_N];
    #pragma unroll
    for (int j = 0; j < OUT_CAP_N; ++j) sacc[j] = 0.f;

    #pragma unroll
    for (int i = 0; i < IN_CAP_SZ; ++i) {
        const float ui = uh[i];
        float m2 = -INFINITY;
        #pragma unroll
        for (int j = 0; j < OUT_CAP_N; ++j)
            m2 = fmaxf(m2, bp[j] + v[j] * ui);
        float d2 = 0.f;
        #pragma unroll
        for (int j = 0; j < OUT_CAP_N; ++j)
            d2 += __expf(bp[j] + v[j] * ui - m2);
        const float i2 = __frcp_rn(d2);
        #pragma unroll
        for (int j = 0; j < OUT_CAP_N; ++j)
            sacc[j] += __expf(bp[j] + v[j] * ui - m2) * i2 * ui;
    }

    #pragma unroll
    for (int j = 0; j < OUT_CAP_N; ++j) {
        float s  = sacc[j];
        float s2 = s * s;
        out[b * OUT_CAP_N + j] = fabsf(s) / (1.f + s2) * s;
    }
}

// ---------------------------------------------------------------------------
extern "C" void kernel_launch(void* const* d_in, const int* in_sizes, int n_in,
                              void* d_out, int out_size, void* d_ws, size_t ws_size,
                              hipStream_t stream) {
    const float* u  = (const float*)d_in[0];   // (8192, 5, 128, 3, 3)
    const float* W  = (const float*)d_in[1];   // (1, 1152, 1)
    const float* bp = (const float*)d_in[2];   // (55, 1)
    float* out   = (float*)d_out;              // (8192, 55, 1)
    float* u_hat = (float*)d_ws;               // ROWS floats (160 KB scratch)

    // 40960 rows / (2 tiles * 16 rows) = 1280 blocks, 8 waves each
    uhat_wmma_kernel<<<ROWS / 32, 256, 0, stream>>>(u, W, u_hat);
    routing_kernel<<<BATCH / 64, 64, 0, stream>>>(u_hat, bp, out);
}